// MaskedPooling_50053548867925
// MI455X (gfx1250) — compile-verified
//
#include <hip/hip_runtime.h>
#include <hip/hip_bf16.h>

// Masked mean pooling: out[b,d] = sum_t keep[b,t]*x[b,t,d] / sum_t keep[b,t]
// x: [32, 4096, 512] f32, mask: [32, 4096] bool (True = excluded)
//
// Bandwidth-bound (256 MiB @ 23.3 TB/s ~= 11 us floor). The MAC chain runs on
// the matrix pipe via v_wmma_f32_16x16x4_f32 (A = keep[t..t+3] replicated over
// all 16 rows per the documented f32 16x4 A layout; B = x tile with mirrored
// K = 2*half+vgpr striping). 1024 blocks x 8 waves stream x once; partials go
// to d_ws and are combined in a fixed order for determinism.

typedef float v2f __attribute__((ext_vector_type(2)));
typedef float v8f __attribute__((ext_vector_type(8)));

#define B_    32
#define T_    4096
#define D_    512
#define SEG   8
#define TSEG  (T_ / SEG)        // 512 t per segment
#define DPW   16                // d-columns per wave (one WMMA tile width)
#define WAVES 8                 // waves per block (256 threads, wave32)
#define DPB   (DPW * WAVES)     // 128 d per block
#define DCH   (D_ / DPB)        // 4 d-chunks

__global__ __launch_bounds__(256)
void masked_pool_partial(const float* __restrict__ x,
                         const unsigned char* __restrict__ mask,
                         float* __restrict__ ws) {
    __shared__ float keepLds[TSEG];

    const int blk = blockIdx.x;
    const int seg = blk % SEG;
    const int dc  = (blk / SEG) % DCH;
    const int b   = blk / (SEG * DCH);

    const int tid   = threadIdx.x;
    const int tbase = seg * TSEG;

    // Stage keep = !mask as f32 into LDS once per block (shared by all 8 waves).
    const unsigned char* mrow = mask + (size_t)b * T_ + tbase;
    for (int t = tid; t < TSEG; t += 256)
        keepLds[t] = (mrow[t] == 0) ? 1.0f : 0.0f;
    __syncthreads();

    const int wave = tid >> 5;
    const int lane = tid & 31;
    const int n    = lane & 15;          // WMMA column within tile
    const int half = lane >> 4;          // lane half selects K = {0,1} vs {2,3}
    const int d    = dc * DPB + wave * DPW + n;

    // Per-lane base pointer into this b / segment / d column.
    const float* xb = x + ((size_t)b * T_ + tbase) * D_ + d;

    // f32 16x4 A layout (ISA 7.12.2): VGPR0 holds K0 (lanes 0-15) / K2 (16-31),
    // VGPR1 holds K1 / K3.  B (4x16) uses the mirrored striping K = 2*half+vgpr.
    const int te = 2 * half;

    v8f c = {0.f, 0.f, 0.f, 0.f, 0.f, 0.f, 0.f, 0.f};

    #pragma unroll 8
    for (int t = 0; t < TSEG; t += 4) {
        v2f a, bt;
        a.x  = keepLds[t + te];
        a.y  = keepLds[t + te + 1];
        bt.x = xb[(size_t)(t + te)     * D_];
        bt.y = xb[(size_t)(t + te + 1) * D_];

        // Unconditional stream-ahead prefetch (global_prefetch_b8 is
        // speculative: past-the-end translations are silently dropped).
        __builtin_prefetch(&xb[(size_t)(t + 16 + te)     * D_], 0, 1);
        __builtin_prefetch(&xb[(size_t)(t + 16 + te + 1) * D_], 0, 1);

        // D = A(16x4) * B(4x16) + C : every output row = sum_k keep[t+k]*x[t+k,d]
        c = __builtin_amdgcn_wmma_f32_16x16x4_f32(
                /*neg_a=*/false, a, /*neg_b=*/false, bt,
                /*c_mod=*/(short)0, c, /*reuse_a=*/false, /*reuse_b=*/false);
    }

    // All rows identical; lane n (lanes 0-15) holds (M=0, N=n) in c[0].
    if (half == 0)
        ws[((size_t)seg * B_ + b) * D_ + d] = c[0];
}

__global__ __launch_bounds__(256)
void masked_pool_finalize(const float* __restrict__ ws,
                          const unsigned char* __restrict__ mask,
                          float* __restrict__ out) {
    __shared__ float red[256];
    const int b   = blockIdx.x;
    const int tid = threadIdx.x;

    // Denominator: count of kept (mask==0) positions in row b.
    const unsigned char* mrow = mask + (size_t)b * T_;
    float local = 0.f;
    for (int t = tid; t < T_; t += 256)
        local += (mrow[t] == 0) ? 1.0f : 0.0f;
    red[tid] = local;
    __syncthreads();
    for (int s = 128; s > 0; s >>= 1) {
        if (tid < s) red[tid] += red[tid + s];
        __syncthreads();
    }
    const float inv = 1.0f / red[0];

    // Fixed-order combine of the SEG partials, then scale.
    for (int d = tid; d < D_; d += 256) {
        float s = 0.f;
        #pragma unroll
        for (int seg = 0; seg < SEG; ++seg)
            s += ws[((size_t)seg * B_ + b) * D_ + d];
        out[(size_t)b * D_ + d] = s * inv;
    }
}

extern "C" void kernel_launch(void* const* d_in, const int* in_sizes, int n_in,
                              void* d_out, int out_size, void* d_ws, size_t ws_size,
                              hipStream_t stream) {
    (void)in_sizes; (void)n_in; (void)out_size; (void)ws_size;
    const float*         x    = (const float*)d_in[0];
    const unsigned char* mask = (const unsigned char*)d_in[1];  // jax bool -> 1 byte
    float*               out  = (float*)d_out;
    float*               ws   = (float*)d_ws;                   // SEG*B*D f32 = 512 KB

    masked_pool_partial<<<B_ * DCH * SEG, 256, 0, stream>>>(x, mask, ws);
    masked_pool_finalize<<<B_, 256, 0, stream>>>(ws, mask, out);
}